// ExpertChoiceRouter_32521492365538
// MI455X (gfx1250) — compile-verified
//
#include <hip/hip_runtime.h>
#include <hip/hip_bf16.h>
#include <math.h>

typedef __attribute__((ext_vector_type(2))) float v2f;
typedef __attribute__((ext_vector_type(8))) float v8f;

#define AUX_COEF 0.001f
#define CHUNK_K  256
#define ROWPAD   260   // 256 + 4 floats pad -> conflict-free 16-row column reads

// ---------------------------------------------------------------------------
// Kernel 1: logits[i] = dot(x[i,:], w) via V_WMMA_F32_16X16X4_F32.
// Block = 128 threads (4 waves) owns one 16-row tile of x.
// Per 256-col chunk: coalesced b128 global->LDS staging, then each wave
// k-splits the chunk (64 cols, 16 WMMAs) reading A-fragments from LDS.
// Partials combined in fixed order -> deterministic. HBM-bound: ~5.5us floor.
// ---------------------------------------------------------------------------
__global__ void __launch_bounds__(128) router_gemv_wmma(
    const float* __restrict__ x, const float* __restrict__ w,
    float* __restrict__ logits, int N, int D)
{
  __shared__ float lds_w[2048];
  __shared__ float tileA[16 * ROWPAD];
  __shared__ float partial[4][16];

  const int tid  = threadIdx.x;      // 0..127
  const int lane = tid & 31;
  const int wave = tid >> 5;         // 0..3
  const int tile = blockIdx.x * 16;  // first of this block's 16 rows
  (void)N;                           // grid sized exactly: tile+15 < N

  // stage w once (D <= 2048, multiple of 4)
  for (int i = tid * 4; i < D; i += 128 * 4)
    *(float4*)&lds_w[i] = *(const float4*)&w[i];

  const int m  = lane & 15;  // A-matrix row (M)
  const int hi = lane >> 4;  // lane half -> K pair {0,1} vs {2,3}

  v8f acc = {};

  for (int kc = 0; kc < D; kc += CHUNK_K) {
    __syncthreads();  // prev chunk's LDS reads done (covers lds_w on iter 0)

    // stage 16 x 256 tile: 8 coalesced float4 loads per thread
    #pragma unroll
    for (int r = 0; r < 8; ++r) {
      const int idx = (r * 128 + tid) * 4;   // linear elem in 16x256 tile
      const int row = idx >> 8;              // /256
      const int col = idx & 255;
      float4 v = *(const float4*)(x + (size_t)(tile + row) * (size_t)D + kc + col);
      *(float4*)&tileA[row * ROWPAD + col] = v;
    }
    __syncthreads();

    // this wave's k-subrange within the chunk
    const int kbase = wave * 64;
    #pragma unroll
    for (int kk = 0; kk < 64; kk += 4) {
      const int c = kbase + kk + 2 * hi;
      // A fragment: v0 = K{k | k+2}, v1 = K{k+1 | k+3} across lane halves
      float2 av = *(const float2*)&tileA[m * ROWPAD + c];
      v2f a; a.x = av.x; a.y = av.y;
      // B fragment: only column N=0 (m==0 lanes) carries w
      float2 wv = *(const float2*)&lds_w[kc + c];
      v2f b; b.x = (m == 0) ? wv.x : 0.0f;
             b.y = (m == 0) ? wv.y : 0.0f;
      acc = __builtin_amdgcn_wmma_f32_16x16x4_f32(
          /*neg_a=*/false, a, /*neg_b=*/false, b,
          /*c_mod=*/(short)0, acc, /*reuse_a=*/false, /*reuse_b=*/false);
    }
  }

  // column 0 of D: m<8 -> VGPR m lane 0 ; m>=8 -> VGPR m-8 lane 16
  if (m == 0) {
    #pragma unroll
    for (int j = 0; j < 8; ++j) partial[wave][8 * hi + j] = acc[j];
  }
  __syncthreads();

  // fixed-order combine of the 4 k-split partials -> deterministic
  if (tid < 16) {
    float s = ((partial[0][tid] + partial[1][tid]) + partial[2][tid]) + partial[3][tid];
    logits[tile + tid] = s;
  }
}

// ---------------------------------------------------------------------------
// Kernel 2: single-workgroup exact top-K (MSB-first radix select on
// order-preserving uint keys), deterministic lowest-index tie-break to match
// jax.lax.top_k, mask + aux-loss output. 64KB of scores is L2-resident.
// ---------------------------------------------------------------------------
__device__ __forceinline__ unsigned key_of(float f, unsigned char active) {
  unsigned bits = __float_as_uint(f);
  unsigned k = (bits & 0x80000000u) ? ~bits : (bits | 0x80000000u);
  return active ? k : 0u;  // inactive tokens sort below everything real
}

__global__ void __launch_bounds__(1024) topk_mask_aux(
    const float* __restrict__ logits, const unsigned char* __restrict__ amask,
    float* __restrict__ out, int N, int K)
{
  __shared__ unsigned int hist[256];
  __shared__ unsigned int scanbuf[1024];
  __shared__ float        fred[1024];
  __shared__ unsigned int sh_pref;
  __shared__ int          sh_rem;

  const int tid = threadIdx.x;
  const int nth = blockDim.x;

  if (tid == 0) { sh_pref = 0u; sh_rem = K; }
  __syncthreads();

  // 4 MSB-first radix passes -> threshold key T, residual tie budget r
  for (int pass = 0; pass < 4; ++pass) {
    const int shift = 24 - pass * 8;
    const unsigned prefMask = (pass == 0) ? 0u : (0xFFFFFFFFu << (32 - pass * 8));
    const unsigned pref = sh_pref;
    for (int b = tid; b < 256; b += nth) hist[b] = 0u;
    __syncthreads();
    for (int i = tid; i < N; i += nth) {
      unsigned k = key_of(logits[i], amask[i]);
      if ((k & prefMask) == (pref & prefMask))
        atomicAdd(&hist[(k >> shift) & 0xFFu], 1u);
    }
    __syncthreads();
    if (tid == 0) {
      int rem = sh_rem;
      unsigned cum = 0u;
      int b = 255;
      for (; b > 0; --b) {
        unsigned c = hist[b];
        if (cum + c >= (unsigned)rem) break;
        cum += c;
      }
      sh_rem  = rem - (int)cum;
      sh_pref = pref | ((unsigned)b << shift);
    }
    __syncthreads();
  }

  const unsigned T = sh_pref;
  const int      r = sh_rem;  // # of key==T elements to keep (lowest idx first)

  const int per  = (N + nth - 1) / nth;
  const int base = tid * per;
  const int end  = (base + per < N) ? (base + per) : N;

  // per-chunk tie counts -> exclusive scan = my first tie's global rank
  int eq = 0;
  for (int i = base; i < end; ++i)
    if (key_of(logits[i], amask[i]) == T) eq++;
  scanbuf[tid] = (unsigned)eq;
  __syncthreads();
  for (int off = 1; off < nth; off <<= 1) {
    unsigned t = (tid >= off) ? scanbuf[tid - off] : 0u;
    __syncthreads();
    scanbuf[tid] += t;
    __syncthreads();
  }
  int rank = (int)scanbuf[tid] - eq;  // exclusive prefix of ties

  float ssum = 0.0f;
  for (int i = base; i < end; ++i) {
    unsigned k = key_of(logits[i], amask[i]);
    bool sel = false;
    if (k > T)       sel = true;
    else if (k == T) { sel = (rank < r); rank++; }
    out[i] = sel ? 1.0f : 0.0f;
    if (sel) ssum += 1.0f / (1.0f + expf(-logits[i]));
  }

  fred[tid] = ssum;
  __syncthreads();
  for (int off = nth >> 1; off > 0; off >>= 1) {
    if (tid < off) fred[tid] += fred[tid + off];
    __syncthreads();
  }
  if (tid == 0) out[N] = -(fred[0] / (float)K) * AUX_COEF;
}

// ---------------------------------------------------------------------------
extern "C" void kernel_launch(void* const* d_in, const int* in_sizes, int n_in,
                              void* d_out, int out_size, void* d_ws, size_t ws_size,
                              hipStream_t stream) {
  const float*         x     = (const float*)d_in[0];
  const unsigned char* amask = (const unsigned char*)d_in[1];  // jnp bool -> 1B
  const float*         w     = (const float*)d_in[2];
  float*               out   = (float*)d_out;
  float*               logit = (float*)d_ws;

  const int N = in_sizes[1];        // B*S = 16384 tokens
  const int D = in_sizes[2];        // d_model = 2048
  int K = (int)((double)N * 0.67);  // matches int(num_active * 0.67)
  if (K < 1) K = 1;

  const int blocks = (N + 15) / 16; // one 16-row tile per block (4 waves)

  router_gemv_wmma<<<blocks, 128, 0, stream>>>(x, w, logit, N, D);
  topk_mask_aux<<<1, 1024, 0, stream>>>(logit, amask, out, N, K);

  (void)n_in; (void)out_size; (void)ws_size;
}